// Attention_32650341384246
// MI455X (gfx1250) — compile-verified
//
#include <hip/hip_runtime.h>
#include <hip/hip_bf16.h>

// ---------------------------------------------------------------------------
// Attention block for gfx1250 (MI455X), fp32 end-to-end using
// V_WMMA_F32_16X16X4_F32, async-to-LDS double-buffered K/V staging in the
// flash-attention kernel, and global_prefetch in the GEMM kernels.
//
// B=2, S=2048, D_MODEL=1024, H=16, D_HEAD=64
// ---------------------------------------------------------------------------

#define B_   2
#define S_   2048
#define DM   1024
#define H_   16
#define DH   64
#define NTOK (B_ * S_)

typedef __attribute__((ext_vector_type(2))) float v2f;
typedef __attribute__((ext_vector_type(8))) float v8f;

// 8-arg pattern for 16x16x{4,32}: (neg_a, A, neg_b, B, c_mod, C, reuse_a, reuse_b)
#define WMMA_F32(a, b, c) \
    __builtin_amdgcn_wmma_f32_16x16x4_f32(false, (a), false, (b), (short)0, (c), false, false)

// ---- async-to-LDS path (gfx1250) -----------------------------------------
// Probe-derived signature (from the round-2 diagnostic): parameter 1 is a
// NON-const addrspace(1) pointer to `int __attribute__((vector_size(16)))`.
#if __has_builtin(__builtin_amdgcn_global_load_async_to_lds_b128) && \
    __has_builtin(__builtin_amdgcn_s_wait_asynccnt)
#define HAS_ASYNC_LDS 1
typedef int v4i_vs __attribute__((vector_size(16)));
#define ASYNC_B128(g, l)                                                     \
    __builtin_amdgcn_global_load_async_to_lds_b128(                          \
        (__attribute__((address_space(1))) v4i_vs*)(g),                      \
        (__attribute__((address_space(3))) v4i_vs*)(l), 0, 0)
#define WAIT_ASYNC(n) __builtin_amdgcn_s_wait_asynccnt(n)
#else
#define HAS_ASYNC_LDS 0
#define ASYNC_B128(g, l)                                                     \
    do {                                                                     \
        *(float4*)(l) = *(const float4*)(g);                                 \
    } while (0)
#define WAIT_ASYNC(n) ((void)0)
#endif

// ---------------------------------------------------------------------------
// Kernel 1: qkv = x @ Wqkv + bqkv, fused RMS-norm + RoPE on q/k heads.
// Grid: (48 col-tiles of 64, 256 row-tiles of 16).  Block: 128 (4 waves).
// ---------------------------------------------------------------------------
__global__ __launch_bounds__(128)
void qkv_rope_kernel(const float* __restrict__ x,       // [NTOK, DM]
                     const float* __restrict__ pos,     // [S_]
                     const float* __restrict__ Wqkv,    // [DM, 3*DM]
                     const float* __restrict__ bqkv,    // [3*DM]
                     const float* __restrict__ q_scale, // [DH]
                     const float* __restrict__ k_scale, // [DH]
                     float* __restrict__ q_ws,          // [B_,H_,S_,DH]
                     float* __restrict__ k_ws,
                     float* __restrict__ v_ws)
{
    __shared__ float tile[16][DH + 1];

    const int lane = threadIdx.x & 31;
    const int wave = threadIdx.x >> 5;
    const int half = lane >> 4;
    const int l16  = lane & 15;

    const int colTile = blockIdx.x;          // 0..47
    const int rowBase = blockIdx.y * 16;
    const int col     = colTile * 64 + wave * 16 + l16;

    v8f acc = {};
    const float* arow = x + (size_t)(rowBase + l16) * DM + half * 2;
    for (int k0 = 0; k0 < DM; k0 += 64) {
        if (k0 + 64 < DM) __builtin_prefetch(arow + k0 + 64, 0, 0);
        #pragma unroll
        for (int kk = 0; kk < 64; kk += 4) {
            const int k = k0 + kk;
            v2f a, b;
            a.x = arow[k];
            a.y = arow[k + 1];
            b.x = Wqkv[(size_t)(k + half * 2)     * (3 * DM) + col];
            b.y = Wqkv[(size_t)(k + half * 2 + 1) * (3 * DM) + col];
            acc = WMMA_F32(a, b, acc);
        }
    }

    const float bias = bqkv[col];
    #pragma unroll
    for (int r = 0; r < 8; ++r)
        tile[r + half * 8][wave * 16 + l16] = acc[r] + bias;
    __syncthreads();

    const int qkvType = colTile >> 4;    // 0=q, 1=k, 2=v
    const int head    = colTile & 15;

    if (threadIdx.x < 16) {
        const int tok = rowBase + threadIdx.x;
        const int bb  = tok / S_;
        const int ss  = tok % S_;

        float vals[DH];
        #pragma unroll
        for (int d = 0; d < DH; ++d) vals[d] = tile[threadIdx.x][d];

        const size_t obase = (((size_t)bb * H_ + head) * S_ + ss) * DH;

        if (qkvType == 2) {
            #pragma unroll
            for (int d = 0; d < DH; ++d) v_ws[obase + d] = vals[d];
        } else {
            float ssum = 0.f;
            #pragma unroll
            for (int d = 0; d < DH; ++d) ssum += vals[d] * vals[d];
            const float inv = rsqrtf(ssum * (1.0f / DH) + 1e-6f);
            const float* sc = (qkvType == 0) ? q_scale : k_scale;
            #pragma unroll
            for (int d = 0; d < DH; ++d) vals[d] = sc[d] * (vals[d] * inv);

            const float p = pos[ss];
            #pragma unroll
            for (int j = 0; j < DH / 2; ++j) {
                const float omega = powf(10000.0f, -2.0f * (float)j / (float)DH);
                const float ang = p * omega;
                const float c = cosf(ang), s = sinf(ang);
                const float x0 = vals[2 * j], x1 = vals[2 * j + 1];
                vals[2 * j]     = x0 * c + x1 * s;
                vals[2 * j + 1] = x1 * c - x0 * s;
            }
            float* outp = ((qkvType == 0) ? q_ws : k_ws);
            #pragma unroll
            for (int d = 0; d < DH; ++d) outp[obase + d] = vals[d];
        }
    }
}

// ---------------------------------------------------------------------------
// Kernel 2: single-pass flash attention, fp32, with cooperative K/V LDS
// staging (async-to-LDS + ASYNCcnt double buffering when available).
// Grid: (B*H = 32, S/64 = 32). Block: 128 (4 waves), wave owns 16 q rows.
// ---------------------------------------------------------------------------
__global__ __launch_bounds__(128)
void flash_attn_kernel(const float* __restrict__ q_ws,
                       const float* __restrict__ k_ws,
                       const float* __restrict__ v_ws,
                       float* __restrict__ o_ws)   // [NTOK, H_*DH] token-major
{
    // rows padded to 68 floats: keeps 16B alignment for B128 async writes and
    // spreads banks (transposed K reads: lane stride = 4 banks).
    __shared__ float kbuf[2][16][68];
    __shared__ float vbuf[2][16][68];
    __shared__ float pbuf[4][16][17];    // per-wave P tile (C->A relayout)

    const int tid  = threadIdx.x;
    const int lane = tid & 31;
    const int wave = tid >> 5;
    const int half = lane >> 4;
    const int l16  = lane & 15;

    const int bh    = blockIdx.x;                 // 0..31
    const int qBase = blockIdx.y * 64 + wave * 16;

    const float* Q = q_ws + (size_t)bh * S_ * DH;
    const float* K = k_ws + (size_t)bh * S_ * DH;
    const float* V = v_ws + (size_t)bh * S_ * DH;

    // cooperative copy assignment: 128 threads cover 16 rows x 64 floats,
    // 8 floats (two B128s) per thread per buffer.
    const int crow = tid >> 3;          // 0..15
    const int cseg = (tid & 7) * 8;     // 0,8,...,56

    // Preload Q as A-fragments: step k covers dims 4k..4k+3
    v2f qa[16];
    {
        const float* qrow = Q + (size_t)(qBase + l16) * DH + half * 2;
        #pragma unroll
        for (int k = 0; k < 16; ++k) { qa[k].x = qrow[4 * k]; qa[k].y = qrow[4 * k + 1]; }
    }

    v8f o0 = {}, o1 = {}, o2 = {}, o3 = {};
    float m_r[8], l_r[8];
    #pragma unroll
    for (int r = 0; r < 8; ++r) { m_r[r] = -3.0e38f; l_r[r] = 0.f; }

    // stage(tile, buf): issue 4 async B128 per thread (K + V halves)
    auto stage = [&](int ktile, int buf) {
        const float* gK = K + (size_t)(ktile + crow) * DH + cseg;
        const float* gV = V + (size_t)(ktile + crow) * DH + cseg;
        float* lK = &kbuf[buf][crow][cseg];
        float* lV = &vbuf[buf][crow][cseg];
        ASYNC_B128(gK,     lK);
        ASYNC_B128(gK + 4, lK + 4);
        ASYNC_B128(gV,     lV);
        ASYNC_B128(gV + 4, lV + 4);
    };

    stage(0, 0);                          // prefetch tile 0

    for (int kt = 0; kt < S_; kt += 16) {
        const int p = (kt >> 4) & 1;

        // prefetch next tile into the other buffer (safe: everyone finished
        // reading it at the end-of-iteration barrier of the previous tile), then
        // wait for the current buffer's 4 async ops (per wave) to land.
        if (kt + 16 < S_) {
            stage(kt + 16, p ^ 1);
            WAIT_ASYNC(4);
        } else {
            WAIT_ASYNC(0);
        }
        __syncthreads();

        // ---- scores: S = Q @ K^T (B frag from LDS: B[d][n] = K[kt+n][d]) --
        v8f s = {};
        #pragma unroll
        for (int k = 0; k < 16; ++k) {
            v2f b;
            b.x = kbuf[p][l16][4 * k + half * 2];
            b.y = kbuf[p][l16][4 * k + half * 2 + 1];
            s = WMMA_F32(qa[k], b, s);
        }

        // ---- online softmax (rows live in 16-lane halves) ----------------
        float alpha[8], pr[8];
        #pragma unroll
        for (int r = 0; r < 8; ++r) {
            float sv = s[r] * 0.125f;              // 1/sqrt(64)
            float mx = sv;
            mx = fmaxf(mx, __shfl_xor(mx, 1, 32));
            mx = fmaxf(mx, __shfl_xor(mx, 2, 32));
            mx = fmaxf(mx, __shfl_xor(mx, 4, 32));
            mx = fmaxf(mx, __shfl_xor(mx, 8, 32));
            const float mnew = fmaxf(m_r[r], mx);
            const float a    = __expf(m_r[r] - mnew);
            const float pv   = __expf(sv - mnew);
            float rs = pv;
            rs += __shfl_xor(rs, 1, 32);
            rs += __shfl_xor(rs, 2, 32);
            rs += __shfl_xor(rs, 4, 32);
            rs += __shfl_xor(rs, 8, 32);
            l_r[r]   = l_r[r] * a + rs;
            m_r[r]   = mnew;
            alpha[r] = a;
            pr[r]    = pv;
        }
        #pragma unroll
        for (int r = 0; r < 8; ++r) {
            o0[r] *= alpha[r]; o1[r] *= alpha[r];
            o2[r] *= alpha[r]; o3[r] *= alpha[r];
        }

        // ---- stage P (C-layout -> LDS); per-wave region, DS in-order -----
        #pragma unroll
        for (int r = 0; r < 8; ++r)
            pbuf[wave][r + half * 8][l16] = pr[r];

        // ---- O += P @ V : A from pbuf, B from vbuf -----------------------
        #pragma unroll
        for (int k = 0; k < 4; ++k) {
            v2f a;
            a.x = pbuf[wave][l16][4 * k + half * 2];
            a.y = pbuf[wave][l16][4 * k + half * 2 + 1];
            const int vr = 4 * k + half * 2;
            v2f b;
            b.x = vbuf[p][vr][ 0 + l16]; b.y = vbuf[p][vr + 1][ 0 + l16]; o0 = WMMA_F32(a, b, o0);
            b.x = vbuf[p][vr][16 + l16]; b.y = vbuf[p][vr + 1][16 + l16]; o1 = WMMA_F32(a, b, o1);
            b.x = vbuf[p][vr][32 + l16]; b.y = vbuf[p][vr + 1][32 + l16]; o2 = WMMA_F32(a, b, o2);
            b.x = vbuf[p][vr][48 + l16]; b.y = vbuf[p][vr + 1][48 + l16]; o3 = WMMA_F32(a, b, o3);
        }
        __syncthreads();   // all waves done reading kbuf/vbuf[p] before reuse
    }

    // ---- normalize and store (token-major, [tok][h*DH + d]) -------------
    const int b = bh / H_;
    const int h = bh % H_;
    #pragma unroll
    for (int r = 0; r < 8; ++r) {
        const float invl = 1.0f / l_r[r];
        const int s_idx  = qBase + r + half * 8;
        const size_t base = ((size_t)(b * S_ + s_idx)) * (H_ * DH) + h * DH + l16;
        o_ws[base + 0 * 16] = o0[r] * invl;
        o_ws[base + 1 * 16] = o1[r] * invl;
        o_ws[base + 2 * 16] = o2[r] * invl;
        o_ws[base + 3 * 16] = o3[r] * invl;
    }
}

// ---------------------------------------------------------------------------
// Kernel 3: out = o_ws @ Wout + bout.
// Grid: (16 col-tiles of 64, 256 row-tiles of 16). Block: 128 (4 waves).
// ---------------------------------------------------------------------------
__global__ __launch_bounds__(128)
void out_proj_kernel(const float* __restrict__ o_ws,   // [NTOK, DM]
                     const float* __restrict__ Wout,   // [DM, DM]
                     const float* __restrict__ bout,   // [DM]
                     float* __restrict__ out)          // [NTOK, DM]
{
    const int lane = threadIdx.x & 31;
    const int wave = threadIdx.x >> 5;
    const int half = lane >> 4;
    const int l16  = lane & 15;

    const int rowBase = blockIdx.y * 16;
    const int col     = blockIdx.x * 64 + wave * 16 + l16;

    v8f acc = {};
    const float* arow = o_ws + (size_t)(rowBase + l16) * DM + half * 2;
    for (int k0 = 0; k0 < DM; k0 += 64) {
        if (k0 + 64 < DM) __builtin_prefetch(arow + k0 + 64, 0, 0);
        #pragma unroll
        for (int kk = 0; kk < 64; kk += 4) {
            const int k = k0 + kk;
            v2f a, b;
            a.x = arow[k];
            a.y = arow[k + 1];
            b.x = Wout[(size_t)(k + half * 2)     * DM + col];
            b.y = Wout[(size_t)(k + half * 2 + 1) * DM + col];
            acc = WMMA_F32(a, b, acc);
        }
    }

    const float bias = bout[col];
    #pragma unroll
    for (int r = 0; r < 8; ++r)
        out[(size_t)(rowBase + r + half * 8) * DM + col] = acc[r] + bias;
}

// ---------------------------------------------------------------------------
extern "C" void kernel_launch(void* const* d_in, const int* in_sizes, int n_in,
                              void* d_out, int out_size, void* d_ws, size_t ws_size,
                              hipStream_t stream)
{
    const float* x       = (const float*)d_in[0];
    const float* pos     = (const float*)d_in[1];
    const float* Wqkv    = (const float*)d_in[2];
    const float* bqkv    = (const float*)d_in[3];
    const float* Wout    = (const float*)d_in[4];
    const float* bout    = (const float*)d_in[5];
    const float* q_scale = (const float*)d_in[6];
    const float* k_scale = (const float*)d_in[7];
    float* out = (float*)d_out;

    const size_t headElems = (size_t)B_ * H_ * S_ * DH;   // 4M floats = 16 MB
    float* q_ws = (float*)d_ws;
    float* k_ws = q_ws + headElems;
    float* v_ws = k_ws + headElems;
    float* o_ws = v_ws + headElems;                        // [NTOK, DM]

    qkv_rope_kernel<<<dim3(48, 256), 128, 0, stream>>>(
        x, pos, Wqkv, bqkv, q_scale, k_scale, q_ws, k_ws, v_ws);

    flash_attn_kernel<<<dim3(B_ * H_, S_ / 64), 128, 0, stream>>>(
        q_ws, k_ws, v_ws, o_ws);

    out_proj_kernel<<<dim3(DM / 64, NTOK / 16), 128, 0, stream>>>(
        o_ws, Wout, bout, out);
}